// GAT_57294863728941
// MI455X (gfx1250) — compile-verified
//
#include <hip/hip_runtime.h>
#include <stdint.h>

#define N_NODES   100000
#define N_EDGES   6400000
#define D_IN      256
#define NEG_SLOPE 0.2f

typedef __attribute__((ext_vector_type(2))) float v2f;
typedef __attribute__((ext_vector_type(8))) float v8f;

// ---------------------------------------------------------------------------
// Kernel 1: fused node transform  xl = x@Wl + bl, xr = x@Wr + br
// GEMM [N x 256] x [256 x 16] (cols 0..1 = Wl, 2..3 = Wr, 4..15 = zero pad)
// via V_WMMA_F32_16X16X4_F32.  One wave = one 16-row tile; 4 waves per block
// share the 16 KB LDS-staged weight panel.
// ---------------------------------------------------------------------------
__global__ __launch_bounds__(128) void gat_node_transform(
    const float* __restrict__ x,
    const float* __restrict__ Wl, const float* __restrict__ bl,
    const float* __restrict__ Wr, const float* __restrict__ br,
    float* __restrict__ xl, float* __restrict__ xr, int nTiles)
{
    __shared__ float lWc[D_IN * 16];     // [k][n] row-major, 16 KB

    const int tid = threadIdx.x;
    for (int i = tid; i < D_IN * 16; i += 128) {
        const int k = i >> 4, n = i & 15;
        float w = 0.0f;
        if (n < 2)      w = Wl[k * 2 + n];
        else if (n < 4) w = Wr[k * 2 + (n - 2)];
        lWc[i] = w;
    }
    __syncthreads();

    const int wave = tid >> 5;
    const int lane = tid & 31;
    const int tile = blockIdx.x * 4 + wave;
    if (tile >= nTiles) return;          // whole wave exits together (EXEC all-1s for WMMA)

    const int row0    = tile * 16;
    const int halfsel = lane >> 4;       // 0 -> K pair {0,1}; 1 -> K pair {2,3}
    const int mn      = lane & 15;       // A: row M; B: col N
    const int koff    = halfsel * 2;
    const float* xrow = x + (size_t)(row0 + mn) * D_IN;

    v8f acc = {};
    #pragma unroll 4
    for (int k = 0; k < D_IN; k += 4) {
        v2f a, b;
        a.x = xrow[k + koff];                     // A 16x4 fp32 fragment
        a.y = xrow[k + koff + 1];
        b.x = lWc[(k + koff) * 16 + mn];          // B 4x16 fp32 fragment (LDS)
        b.y = lWc[(k + koff + 1) * 16 + mn];
        acc = __builtin_amdgcn_wmma_f32_16x16x4_f32(
                  false, a, false, b, (short)0, acc, false, false);
    }

    // C/D layout: VGPR v -> M = v (lanes 0-15) / v+8 (lanes 16-31), N = lane&15
    if (mn < 4) {
        const int col = mn;
        #pragma unroll
        for (int v = 0; v < 8; ++v) {
            const int row = row0 + v + halfsel * 8;
            const float val = acc[v];
            if (col < 2) xl[row * 2 + col]       = val + bl[col];
            else         xr[row * 2 + (col - 2)] = val + br[col - 2];
        }
    }
}

// ---------------------------------------------------------------------------
// Init: amax = -inf, denom = 0, out = bias (broadcast)
// ---------------------------------------------------------------------------
__global__ void gat_init(float* __restrict__ amax, float* __restrict__ denom,
                         float* __restrict__ out, const float* __restrict__ bias, int n)
{
    const int i = blockIdx.x * blockDim.x + threadIdx.x;
    if (i >= n) return;
    ((unsigned int*)amax)[i] = 0xFF800000u;   // -inf
    denom[i] = 0.0f;
    out[2 * i + 0] = bias[0];
    out[2 * i + 1] = bias[1];
}

// float atomic max via monotone int mapping (works across sign mix)
__device__ __forceinline__ void atomicMaxF32(float* addr, float val)
{
    if (val >= 0.0f) atomicMax((int*)addr, __float_as_int(val));
    else             atomicMin((unsigned int*)addr, __float_as_uint(val));
}

// ---------------------------------------------------------------------------
// Edge pass 1: alpha_e = leaky(xl[s] + xr[t] + e*We) . att ; segment max
// ---------------------------------------------------------------------------
__global__ void gat_edge_score(const long long* __restrict__ ei,
                               const float* __restrict__ ea,
                               const float* __restrict__ xl,
                               const float* __restrict__ xr,
                               const float* __restrict__ We,
                               const float* __restrict__ att,
                               float* __restrict__ alpha,
                               float* __restrict__ amax)
{
    const int e = blockIdx.x * blockDim.x + threadIdx.x;
    if (e >= N_EDGES) return;
    const long long s = ei[e];
    const long long t = ei[(size_t)N_EDGES + e];
    const float w = ea[e];
    float m0 = xl[s * 2 + 0] + xr[t * 2 + 0] + w * We[0];
    float m1 = xl[s * 2 + 1] + xr[t * 2 + 1] + w * We[1];
    m0 = (m0 > 0.0f) ? m0 : NEG_SLOPE * m0;
    m1 = (m1 > 0.0f) ? m1 : NEG_SLOPE * m1;
    const float al = m0 * att[0] + m1 * att[1];
    alpha[e] = al;
    atomicMaxF32(&amax[t], al);
}

// ---------------------------------------------------------------------------
// Edge pass 2: ex = exp(alpha - amax[t]) ; segment sum of ex
// ---------------------------------------------------------------------------
__global__ void gat_edge_exp(const long long* __restrict__ ei,
                             float* __restrict__ alpha,
                             const float* __restrict__ amax,
                             float* __restrict__ denom)
{
    const int e = blockIdx.x * blockDim.x + threadIdx.x;
    if (e >= N_EDGES) return;
    const long long t = ei[(size_t)N_EDGES + e];
    const float ex = __expf(alpha[e] - amax[t]);
    alpha[e] = ex;
    atomicAdd(&denom[t], ex);
}

// ---------------------------------------------------------------------------
// Edge pass 3: out[t] += (ex / (denom[t] + 1e-16)) * xl[s]
// ---------------------------------------------------------------------------
__global__ void gat_edge_aggr(const long long* __restrict__ ei,
                              const float* __restrict__ alpha,
                              const float* __restrict__ denom,
                              const float* __restrict__ xl,
                              float* __restrict__ out)
{
    const int e = blockIdx.x * blockDim.x + threadIdx.x;
    if (e >= N_EDGES) return;
    const long long s = ei[e];
    const long long t = ei[(size_t)N_EDGES + e];
    const float a = alpha[e] / (denom[t] + 1e-16f);
    atomicAdd(&out[t * 2 + 0], a * xl[s * 2 + 0]);
    atomicAdd(&out[t * 2 + 1], a * xl[s * 2 + 1]);
}

// ---------------------------------------------------------------------------
extern "C" void kernel_launch(void* const* d_in, const int* in_sizes, int n_in,
                              void* d_out, int out_size, void* d_ws, size_t ws_size,
                              hipStream_t stream)
{
    (void)in_sizes; (void)n_in; (void)out_size; (void)ws_size;
    const float*     x    = (const float*)d_in[0];
    const long long* ei   = (const long long*)d_in[1];   // int64 [2, E]
    const float*     ea   = (const float*)d_in[2];
    const float*     Wl   = (const float*)d_in[3];
    const float*     bl   = (const float*)d_in[4];
    const float*     Wr   = (const float*)d_in[5];
    const float*     br   = (const float*)d_in[6];
    const float*     We   = (const float*)d_in[7];
    const float*     att  = (const float*)d_in[8];
    const float*     bias = (const float*)d_in[9];
    float* out = (float*)d_out;

    char* ws = (char*)d_ws;
    float* xl    = (float*)(ws);                 //  800 KB
    float* xr    = (float*)(ws + 800000);        //  800 KB
    float* amax  = (float*)(ws + 1600000);       //  400 KB
    float* denom = (float*)(ws + 2000000);       //  400 KB
    float* alpha = (float*)(ws + 2400000);       // 25.6 MB

    gat_init<<<(N_NODES + 255) / 256, 256, 0, stream>>>(amax, denom, out, bias, N_NODES);

    const int nTiles = (N_NODES + 15) / 16;      // 6250
    gat_node_transform<<<(nTiles + 3) / 4, 128, 0, stream>>>(x, Wl, bl, Wr, br, xl, xr, nTiles);

    const int eBlocks = (N_EDGES + 255) / 256;
    gat_edge_score<<<eBlocks, 256, 0, stream>>>(ei, ea, xl, xr, We, att, alpha, amax);
    gat_edge_exp  <<<eBlocks, 256, 0, stream>>>(ei, alpha, amax, denom);
    gat_edge_aggr <<<eBlocks, 256, 0, stream>>>(ei, alpha, denom, xl, out);
}